// network_88287347736707
// MI455X (gfx1250) — compile-verified
//
#include <hip/hip_runtime.h>
#include <hip/hip_bf16.h>

// ---------------- problem constants (fixed by the reference) ----------------
#define NNODE 50000
#define NEDGE 800000
#define HE    256      // H * E_HEAD
#define EHD   64       // E_HEAD
#define NH    4        // heads
#define NGRP  500
#define EPB   16       // edges per block in the accumulate kernel

typedef __attribute__((ext_vector_type(2))) float v2f;
typedef __attribute__((ext_vector_type(8))) float v8f;

// Monotonic float<->uint encoding so atomicMax(unsigned) implements float max.
// enc is order-preserving; 0 encodes "below -inf", so zero-init works.
__device__ __forceinline__ unsigned fenc(float f) {
  unsigned u = __float_as_uint(f);
  return (u & 0x80000000u) ? ~u : (u | 0x80000000u);
}
__device__ __forceinline__ float fdec(unsigned e) {
  unsigned u = (e & 0x80000000u) ? (e & 0x7FFFFFFFu) : ~e;
  return __uint_as_float(u);
}

// ---------------- fills ----------------
__global__ void fillf_kernel(float* __restrict__ p, float v, size_t n) {
  size_t i = (size_t)blockIdx.x * blockDim.x + threadIdx.x;
  if (i < n) p[i] = v;
}
__global__ void fillu_kernel(unsigned* __restrict__ p, unsigned v, size_t n) {
  size_t i = (size_t)blockIdx.x * blockDim.x + threadIdx.x;
  if (i < n) p[i] = v;
}
__global__ void filli_kernel(int* __restrict__ p, int v, size_t n) {
  size_t i = (size_t)blockIdx.x * blockDim.x + threadIdx.x;
  if (i < n) p[i] = v;
}

// ---------------- Tensor Data Mover: 2D weight-panel load to LDS ------------
#if __has_builtin(__builtin_amdgcn_tensor_load_to_lds) && \
    __has_builtin(__builtin_amdgcn_s_wait_tensorcnt)
#define USE_TDM 1
typedef unsigned tdm_u32x4 __attribute__((ext_vector_type(4)));
typedef int      tdm_i32x4 __attribute__((ext_vector_type(4)));
typedef int      tdm_i32x8 __attribute__((ext_vector_type(8)));

// Load a [rows x row_elems] f32 tile (row stride = row_stride elems) from
// gaddr into LDS at lds_byte_addr. D# packing per CDNA5 ISA section 8.3/8.4.
// Toolchain here exposes the 6-arg builtin:
//   (u32x4 g0, i32x8 g1, i32x4 g2, i32x4 g3, i32x8 extra, i32 cpol)
__device__ __forceinline__ void tdm_load_2d_f32(unsigned lds_byte_addr,
                                                const float* gaddr,
                                                int rows, int row_elems,
                                                int tensor_row_len,
                                                int row_stride) {
  unsigned long long ga = (unsigned long long)(size_t)gaddr;
  tdm_u32x4 g0;
  g0[0] = 1u;                                            // count=1 (valid), user mode
  g0[1] = lds_byte_addr;                                 // lds_addr
  g0[2] = (unsigned)(ga & 0xFFFFFFFFu);                  // global_addr[31:0]
  g0[3] = (unsigned)((ga >> 32) & 0x01FFFFFFu) | (2u << 30); // addr[56:32] | type=2
  tdm_i32x8 g1;
  g1[0] = (int)(2u << 16);                               // data_size=4B; mask=0
  g1[1] = (int)(((unsigned)tensor_row_len & 0xFFFFu) << 16);   // tensor_dim0 lo16
  g1[2] = (int)((((unsigned)tensor_row_len >> 16) & 0xFFFFu) |
                (((unsigned)rows & 0xFFFFu) << 16));     // tensor_dim0 hi | dim1 lo
  g1[3] = (int)((((unsigned)rows >> 16) & 0xFFFFu) |
                (((unsigned)row_elems & 0xFFFFu) << 16)); // dim1 hi | tile_dim0
  g1[4] = (int)((unsigned)rows & 0xFFFFu);               // tile_dim1 (tile_dim2=0)
  g1[5] = row_stride;                                    // tensor_dim0_stride[31:0]
  g1[6] = 0;                                             // stride hi / dim1_stride
  g1[7] = 0;
  tdm_i32x4 z4 = {0, 0, 0, 0};
  tdm_i32x8 z8 = {0, 0, 0, 0, 0, 0, 0, 0};
  __builtin_amdgcn_tensor_load_to_lds(g0, g1, z4, z4, z8, 0);
}
#else
#define USE_TDM 0
#endif

// ---------------- WMMA f32 GEMM: C[M,Nc] = A[M,K] @ W[K,Nc] + bias ----------
// Block = 128 threads (4 waves). Block covers a 64-row x 64-col C panel:
// wave w -> 16 rows, all 4 column sub-tiles (A fragment reused 4x per K-step).
// The K x 64 weight panel is staged in LDS (TDM when available).
// Requires K%4==0, Nc%64==0.
template <int RELU>
__global__ void gemm_wmma_f32(const float* __restrict__ A,
                              const float* __restrict__ W,
                              const float* __restrict__ bias,
                              float* __restrict__ C,
                              int M, int K, int Nc) {
  extern __shared__ float sW[];                 // K*64 floats
  const int n0 = blockIdx.x * 64;
#if USE_TDM
  if (threadIdx.x < 32) {                       // one wave drives the DMA
    tdm_load_2d_f32((unsigned)(size_t)sW, W + n0, K, 64, Nc, Nc);
    __builtin_amdgcn_s_wait_tensorcnt(0);
  }
  __syncthreads();
#else
  for (int i = threadIdx.x; i < K * 64; i += blockDim.x)
    sW[i] = W[(size_t)(i >> 6) * Nc + n0 + (i & 63)];
  __syncthreads();
#endif
  const int lane = threadIdx.x & 31;
  const int wv   = threadIdx.x >> 5;
  const int m0   = (blockIdx.y * 4 + wv) * 16;
  if (m0 >= M) return;                          // wave-uniform; after barrier
  const int row = lane & 15;
  const int hi  = lane >> 4;                    // 0 or 1
  const int col = lane & 15;
  const float* __restrict__ arow = A + (size_t)(m0 + row) * K;
  v8f acc0 = {}, acc1 = {}, acc2 = {}, acc3 = {};
  for (int kk = 0; kk < K; kk += 4) {
    const int ka = kk + 2 * hi;
    v2f a, b;
    a.x = arow[ka];
    a.y = arow[ka + 1];
    b.x = sW[ka * 64 + col];        b.y = sW[(ka + 1) * 64 + col];
    acc0 = __builtin_amdgcn_wmma_f32_16x16x4_f32(false, a, false, b, (short)0, acc0, false, false);
    b.x = sW[ka * 64 + 16 + col];   b.y = sW[(ka + 1) * 64 + 16 + col];
    acc1 = __builtin_amdgcn_wmma_f32_16x16x4_f32(false, a, false, b, (short)0, acc1, false, false);
    b.x = sW[ka * 64 + 32 + col];   b.y = sW[(ka + 1) * 64 + 32 + col];
    acc2 = __builtin_amdgcn_wmma_f32_16x16x4_f32(false, a, false, b, (short)0, acc2, false, false);
    b.x = sW[ka * 64 + 48 + col];   b.y = sW[(ka + 1) * 64 + 48 + col];
    acc3 = __builtin_amdgcn_wmma_f32_16x16x4_f32(false, a, false, b, (short)0, acc3, false, false);
  }
  const float bv0 = bias ? bias[n0 + col]      : 0.0f;
  const float bv1 = bias ? bias[n0 + 16 + col] : 0.0f;
  const float bv2 = bias ? bias[n0 + 32 + col] : 0.0f;
  const float bv3 = bias ? bias[n0 + 48 + col] : 0.0f;
#pragma unroll
  for (int r = 0; r < 8; ++r) {
    float* __restrict__ crow = C + (size_t)(m0 + r + 8 * hi) * Nc + n0;
    float v0 = acc0[r] + bv0, v1 = acc1[r] + bv1;
    float v2 = acc2[r] + bv2, v3 = acc3[r] + bv3;
    if (RELU) {
      v0 = fmaxf(v0, 0.0f); v1 = fmaxf(v1, 0.0f);
      v2 = fmaxf(v2, 0.0f); v3 = fmaxf(v3, 0.0f);
    }
    crow[col] = v0; crow[16 + col] = v1; crow[32 + col] = v2; crow[48 + col] = v3;
  }
}

// ---------------- edge pass B: alpha + segment max ----------------
// one thread per (edge, head); We staged transposed in LDS: sWeT[c*16+j]
__global__ void edge_alpha_kernel(const float* __restrict__ eattr,
                                  const float* __restrict__ We,
                                  const float* __restrict__ q,
                                  const float* __restrict__ k,
                                  const int* __restrict__ src,
                                  const int* __restrict__ tgt,
                                  float* __restrict__ alpha,
                                  unsigned* __restrict__ m_enc,
                                  int E4) {
  __shared__ float sWeT[16 * HE];
  for (int i = threadIdx.x; i < 16 * HE; i += blockDim.x)
    sWeT[i] = We[(size_t)(i & 15) * HE + (i >> 4)];   // transpose on load
  __syncthreads();
  const int gid = blockIdx.x * blockDim.x + threadIdx.x;
  if (gid >= E4) return;
  const int e = gid >> 2, h = gid & 3;
  const int s = src[e], t = tgt[e];
  float ea[16];
#pragma unroll
  for (int j = 0; j < 16; ++j) ea[j] = eattr[(size_t)e * 16 + j];
  const float* __restrict__ qp = q + (size_t)t * HE + h * EHD;
  const float* __restrict__ kp = k + (size_t)s * HE + h * EHD;
  float acc = 0.0f;
  for (int d = 0; d < EHD; ++d) {
    const float* __restrict__ wc = sWeT + (h * EHD + d) * 16;
    float ev = 0.0f;
#pragma unroll
    for (int j = 0; j < 16; ++j) ev += ea[j] * wc[j];
    acc += qp[d] * (kp[d] + ev);
  }
  const float av = acc * 0.125f;       // 1/sqrt(64)
  alpha[gid] = av;
  atomicMax(&m_enc[t * NH + h], fenc(av));
}

// ---------------- edge pass C: exp + segment sum ----------------
__global__ void edge_exp_kernel(const int* __restrict__ tgt,
                                const unsigned* __restrict__ m_enc,
                                float* __restrict__ alpha,
                                float* __restrict__ ssum, int E4) {
  const int gid = blockIdx.x * blockDim.x + threadIdx.x;
  if (gid >= E4) return;
  const int e = gid >> 2, h = gid & 3;
  const int t = tgt[e];
  const float a = expf(alpha[gid] - fdec(m_enc[t * NH + h]));
  alpha[gid] = a;
  atomicAdd(&ssum[t * NH + h], a);
}

// ---------------- edge pass D: normalized scatter-accumulate ----------------
// block = 256 threads (one per output dim), EPB edges per block
__global__ void edge_accum_kernel(const float* __restrict__ eattr,
                                  const float* __restrict__ We,
                                  const float* __restrict__ v,
                                  const int* __restrict__ src,
                                  const int* __restrict__ tgt,
                                  const float* __restrict__ alpha,
                                  const float* __restrict__ ssum,
                                  float* __restrict__ att, int E) {
  __shared__ float sWeT[16 * HE];
  __shared__ float sEA[EPB * 16];
  for (int i = threadIdx.x; i < 16 * HE; i += blockDim.x)
    sWeT[i] = We[(size_t)(i & 15) * HE + (i >> 4)];
  const int e0 = blockIdx.x * EPB;
  {
    const int i = threadIdx.x;          // 256 == EPB*16
    const int ee = e0 + (i >> 4);
    sEA[i] = (ee < E) ? eattr[(size_t)ee * 16 + (i & 15)] : 0.0f;
  }
  __syncthreads();
  const int d = threadIdx.x;
  const int h = d >> 6;
  const float* __restrict__ wc = sWeT + d * 16;
  for (int i = 0; i < EPB; ++i) {
    const int e = e0 + i;
    if (e < E) {
      const int sn = src[e], t = tgt[e];
      const float w = alpha[(size_t)e * NH + h] / ssum[t * NH + h];
      float ev = 0.0f;
#pragma unroll
      for (int j = 0; j < 16; ++j) ev += sEA[i * 16 + j] * wc[j];
      atomicAdd(&att[(size_t)t * HE + d],
                w * (v[(size_t)sn * HE + d] + ev));
    }
  }
}

// ---------------- node pass: beta gate + blend ----------------
__global__ void node_beta_kernel(const float* __restrict__ att,
                                 const float* __restrict__ xr,
                                 const float* __restrict__ Wb,
                                 float* __restrict__ hnew) {
  const int n = blockIdx.x;
  const int d = threadIdx.x;            // 0..255
  const float o  = att[(size_t)n * HE + d];
  const float xv = xr[(size_t)n * HE + d];
  __shared__ float red[HE];
  red[d] = o * Wb[d] + xv * Wb[HE + d] + (o - xv) * Wb[2 * HE + d];
  __syncthreads();
  for (int off = HE / 2; off > 0; off >>= 1) {
    if (d < off) red[d] += red[d + off];
    __syncthreads();
  }
  const float beta = 1.0f / (1.0f + expf(-red[0]));
  hnew[(size_t)n * HE + d] = beta * xv + (1.0f - beta) * o;
}

// ---------------- batchnorm ----------------
__global__ void bn_stats_kernel(const float* __restrict__ h,
                                float* __restrict__ sum,
                                float* __restrict__ sumsq, int M) {
  const int c  = threadIdx.x & 63;
  const int rl = threadIdx.x >> 6;      // 0..3
  float s0 = 0.0f, s1 = 0.0f;
  for (int r = blockIdx.x * 4 + rl; r < M; r += gridDim.x * 4) {
    const float v = h[(size_t)r * 64 + c];
    s0 += v; s1 += v * v;
  }
  __shared__ float r0[256], r1[256];
  r0[threadIdx.x] = s0; r1[threadIdx.x] = s1;
  __syncthreads();
  if (rl == 0) {
    s0 = r0[c] + r0[64 + c] + r0[128 + c] + r0[192 + c];
    s1 = r1[c] + r1[64 + c] + r1[128 + c] + r1[192 + c];
    atomicAdd(&sum[c], s0);
    atomicAdd(&sumsq[c], s1);
  }
}

__global__ void bn_apply_kernel(float* __restrict__ h,
                                const float* __restrict__ sum,
                                const float* __restrict__ sumsq,
                                const float* __restrict__ g,
                                const float* __restrict__ b, int M) {
  const size_t gid = (size_t)blockIdx.x * blockDim.x + threadIdx.x;
  if (gid >= (size_t)M * 64) return;
  const int c = (int)(gid & 63);
  const float mu  = sum[c] / (float)M;
  const float var = sumsq[c] / (float)M - mu * mu;
  h[gid] = (h[gid] - mu) * rsqrtf(var + 1e-5f) * g[c] + b[c];
}

// ---------------- pooling ----------------
__global__ void counts_kernel(const int* __restrict__ batch,
                              int* __restrict__ counts, int M) {
  const int n = blockIdx.x * blockDim.x + threadIdx.x;
  if (n < M) atomicAdd(&counts[batch[n]], 1);
}

__global__ void pool_kernel(const float* __restrict__ h,
                            const int* __restrict__ batch,
                            unsigned* __restrict__ gmax,
                            float* __restrict__ gsum, int M) {
  const size_t gid = (size_t)blockIdx.x * blockDim.x + threadIdx.x;
  if (gid >= (size_t)M * 64) return;
  const int n = (int)(gid >> 6), c = (int)(gid & 63);
  const int g = batch[n];
  const float v = h[gid];
  atomicMax(&gmax[g * 64 + c], fenc(v));
  atomicAdd(&gsum[g * 64 + c], v);
}

__global__ void pool_finalize_kernel(unsigned* __restrict__ gmax,
                                     float* __restrict__ gsum,
                                     const int* __restrict__ counts,
                                     float* __restrict__ rep) {
  const int gid = blockIdx.x * blockDim.x + threadIdx.x;
  if (gid >= NGRP * 64) return;
  const int g = gid >> 6, c = gid & 63;
  const int cnt = counts[g];
  const float gm   = (cnt > 0) ? fdec(gmax[gid]) : 0.0f;
  const float mean = gsum[gid] / fmaxf((float)cnt, 1.0f);
  rep[g * 128 + c]      += gm;
  rep[g * 128 + 64 + c] += mean;
  gmax[gid] = 0u;                        // reset for next layer
  gsum[gid] = 0.0f;
}

// ---------------- small dense (final MLP, M=500) ----------------
template <int RELU>
__global__ void dense_kernel(const float* __restrict__ A,
                             const float* __restrict__ W,
                             const float* __restrict__ b,
                             float* __restrict__ C, int M, int K, int Nc) {
  const int gid = blockIdx.x * blockDim.x + threadIdx.x;
  if (gid >= M * Nc) return;
  const int m = gid / Nc, n = gid % Nc;
  float acc = b ? b[n] : 0.0f;
  for (int kk = 0; kk < K; ++kk)
    acc += A[(size_t)m * K + kk] * W[(size_t)kk * Nc + n];
  if (RELU) acc = fmaxf(acc, 0.0f);
  C[gid] = acc;
}

// ---------------- host orchestration ----------------
struct ConvP {
  const float *Wq, *bq, *Wk, *bk, *Wv, *bv, *We, *Wskip, *bskip, *Wbeta;
};
static ConvP grab_conv(void* const* d, int base) {
  ConvP p;
  p.Wq    = (const float*)d[base + 0];
  p.bq    = (const float*)d[base + 1];
  p.Wk    = (const float*)d[base + 2];
  p.bk    = (const float*)d[base + 3];
  p.Wv    = (const float*)d[base + 4];
  p.bv    = (const float*)d[base + 5];
  p.We    = (const float*)d[base + 6];
  p.Wskip = (const float*)d[base + 7];
  p.bskip = (const float*)d[base + 8];
  p.Wbeta = (const float*)d[base + 9];
  return p;
}
static inline int gdiv(size_t n, int b) { return (int)((n + (size_t)b - 1) / b); }

extern "C" void kernel_launch(void* const* d_in, const int* in_sizes, int n_in,
                              void* d_out, int out_size, void* d_ws, size_t ws_size,
                              hipStream_t stream) {
  const float* x     = (const float*)d_in[0];
  const float* eattr = (const float*)d_in[1];
  const int*   eidx  = (const int*)d_in[2];   // [2,E] int32
  const int*   src   = eidx;
  const int*   tgt   = eidx + NEDGE;
  const int*   batch = (const int*)d_in[3];

  ConvP conv0 = grab_conv(d_in, 4);
  const float* lin0_W = (const float*)d_in[14];
  const float* lin0_b = (const float*)d_in[15];
  const float* bn0_g  = (const float*)d_in[16];
  const float* bn0_b  = (const float*)d_in[17];
  ConvP convs[3] = { grab_conv(d_in, 18), grab_conv(d_in, 28), grab_conv(d_in, 38) };
  const float *lins_W[3], *lins_b[3], *bns_g[3], *bns_b[3];
  for (int i = 0; i < 3; ++i) {
    lins_W[i] = (const float*)d_in[48 + i];
    lins_b[i] = (const float*)d_in[51 + i];
    bns_g[i]  = (const float*)d_in[54 + i];
    bns_b[i]  = (const float*)d_in[57 + i];
  }
  const float* l1_W = (const float*)d_in[60];
  const float* l1_b = (const float*)d_in[61];
  const float* l2_W = (const float*)d_in[62];
  const float* l2_b = (const float*)d_in[63];
  const float* l3_W = (const float*)d_in[64];
  const float* l3_b = (const float*)d_in[65];

  // ---- workspace carve (floats) ----
  float* ws = (float*)d_ws;
  size_t off = 0;
  auto alloc = [&](size_t n) { float* p = ws + off; off += n; return p; };
  float* q    = alloc((size_t)NNODE * HE);
  float* kbuf = alloc((size_t)NNODE * HE);
  float* vbuf = alloc((size_t)NNODE * HE);
  float* xr   = alloc((size_t)NNODE * HE);
  float* att  = alloc((size_t)NNODE * HE);
  float* hnew = alloc((size_t)NNODE * HE);
  float* h64  = alloc((size_t)NNODE * 64);
  float* alpha = alloc((size_t)NEDGE * NH);
  unsigned* m_enc = (unsigned*)alloc((size_t)NNODE * NH);
  float* ssum  = alloc((size_t)NNODE * NH);
  float* bn_s  = alloc(64);
  float* bn_q  = alloc(64);
  unsigned* gmax = (unsigned*)alloc((size_t)NGRP * 64);
  float* gsum  = alloc((size_t)NGRP * 64);
  int*   counts = (int*)alloc(NGRP);
  float* rep = alloc((size_t)NGRP * 128);
  float* g1  = alloc((size_t)NGRP * 256);
  float* g2  = alloc((size_t)NGRP * 128);
  (void)ws_size; (void)in_sizes; (void)n_in; (void)out_size;

  const int T = 256;
  // persistent accumulators
  fillu_kernel<<<gdiv(NGRP * 64, T), T, 0, stream>>>(gmax, 0u, NGRP * 64);
  fillf_kernel<<<gdiv(NGRP * 64, T), T, 0, stream>>>(gsum, 0.0f, NGRP * 64);
  filli_kernel<<<gdiv(NGRP, T), T, 0, stream>>>(counts, 0, NGRP);
  fillf_kernel<<<gdiv(NGRP * 128, T), T, 0, stream>>>(rep, 0.0f, NGRP * 128);
  counts_kernel<<<gdiv(NNODE, T), T, 0, stream>>>(batch, counts, NNODE);

  const int E4 = NEDGE * NH;
  const int MT = (NNODE / 16 + 3) / 4;         // 782 row-panel blocks

  auto run_conv = [&](const float* hin, int K, const ConvP& p) {
    dim3 gg(HE / 64, MT);                      // (4, 782), 4 waves/block
    const size_t sh = (size_t)K * 64 * sizeof(float);
    gemm_wmma_f32<0><<<gg, 128, sh, stream>>>(hin, p.Wq, p.bq, q,    NNODE, K, HE);
    gemm_wmma_f32<0><<<gg, 128, sh, stream>>>(hin, p.Wk, p.bk, kbuf, NNODE, K, HE);
    gemm_wmma_f32<0><<<gg, 128, sh, stream>>>(hin, p.Wv, p.bv, vbuf, NNODE, K, HE);
    gemm_wmma_f32<0><<<gg, 128, sh, stream>>>(hin, p.Wskip, p.bskip, xr, NNODE, K, HE);
    fillu_kernel<<<gdiv((size_t)NNODE * NH, T), T, 0, stream>>>(m_enc, 0u, (size_t)NNODE * NH);
    fillf_kernel<<<gdiv((size_t)NNODE * NH, T), T, 0, stream>>>(ssum, 0.0f, (size_t)NNODE * NH);
    fillf_kernel<<<gdiv((size_t)NNODE * HE, T), T, 0, stream>>>(att, 0.0f, (size_t)NNODE * HE);
    edge_alpha_kernel<<<E4 / T, T, 0, stream>>>(eattr, p.We, q, kbuf, src, tgt, alpha, m_enc, E4);
    edge_exp_kernel<<<E4 / T, T, 0, stream>>>(tgt, m_enc, alpha, ssum, E4);
    edge_accum_kernel<<<NEDGE / EPB, T, 0, stream>>>(eattr, p.We, vbuf, src, tgt, alpha, ssum, att, NEDGE);
    node_beta_kernel<<<NNODE, HE, 0, stream>>>(att, xr, p.Wbeta, hnew);
  };

  auto run_lin_bn = [&](const float* W, const float* b, const float* g, const float* bb) {
    dim3 gg(1, MT);                            // Nc=64 -> one 64-col panel
    const size_t sh = (size_t)HE * 64 * sizeof(float);
    gemm_wmma_f32<1><<<gg, 128, sh, stream>>>(hnew, W, b, h64, NNODE, HE, 64);
    fillf_kernel<<<1, 64, 0, stream>>>(bn_s, 0.0f, 64);
    fillf_kernel<<<1, 64, 0, stream>>>(bn_q, 0.0f, 64);
    bn_stats_kernel<<<200, 256, 0, stream>>>(h64, bn_s, bn_q, NNODE);
    bn_apply_kernel<<<gdiv((size_t)NNODE * 64, T), T, 0, stream>>>(h64, bn_s, bn_q, g, bb, NNODE);
  };

  // layer 0
  run_conv(x, 128, conv0);
  run_lin_bn(lin0_W, lin0_b, bn0_g, bn0_b);
  // layers 1..3 with pooling
  for (int i = 0; i < 3; ++i) {
    run_conv(h64, 64, convs[i]);
    run_lin_bn(lins_W[i], lins_b[i], bns_g[i], bns_b[i]);
    pool_kernel<<<gdiv((size_t)NNODE * 64, T), T, 0, stream>>>(h64, batch, gmax, gsum, NNODE);
    pool_finalize_kernel<<<gdiv(NGRP * 64, T), T, 0, stream>>>(gmax, gsum, counts, rep);
  }
  // final MLP on [500,128]
  dense_kernel<1><<<gdiv(NGRP * 256, T), T, 0, stream>>>(rep, l1_W, l1_b, g1, NGRP, 128, 256);
  dense_kernel<1><<<gdiv(NGRP * 128, T), T, 0, stream>>>(g1, l2_W, l2_b, g2, NGRP, 256, 128);
  dense_kernel<0><<<gdiv(NGRP, T), T, 0, stream>>>(g2, l3_W, l3_b, (float*)d_out, NGRP, 128, 1);
}